// FLAME_1623497637946
// MI455X (gfx1250) — compile-verified
//
#include <hip/hip_runtime.h>

// ---------------------------------------------------------------------------
// FLAME forward on gfx1250: fp32 WMMA (V_WMMA_F32_16X16X4_F32) for both GEMMs.
// B=512 batches, V=5023 vertices, betas K=350, pose K=36.
// ---------------------------------------------------------------------------

#define V_NUM 5023
#define NV3   15069          // V*3  (GEMM N dimension)
#define LK    350            // shape+expression K
#define BATCH 512

typedef __attribute__((ext_vector_type(2))) float v2f;
typedef __attribute__((ext_vector_type(8))) float v8f;

// ===========================================================================
// Kernel 1: v_shaped[b,n] = v_template[n] + sum_l betas[b,l]*SD[n,l]
//                         + eyelid[b,0]*l_eye[n] + eyelid[b,1]*r_eye[n]
// WG = 8 waves; tile = 16 batches x 128 columns; each wave one 16x16 tile.
// ===========================================================================
#define K1_STRIDE 356   // LDS row stride (floats): m*356 mod 64 banks -> spaced by 4, conflict-free
__global__ __launch_bounds__(256) void k1_shape(
    const float* __restrict__ sp,     // [B,300]
    const float* __restrict__ ep,     // [B,50]
    const float* __restrict__ eyelid, // [B,2]
    const float* __restrict__ vt,     // [NV3]
    const float* __restrict__ SD,     // [NV3,350] (shapedirs, row-major over l)
    const float* __restrict__ le,     // [NV3]
    const float* __restrict__ re,     // [NV3]
    float* __restrict__ vs)           // [B,NV3]
{
    __shared__ float sB[16 * K1_STRIDE];
    __shared__ float sEl[16][2];
    const int tid = threadIdx.x;
    const int b0  = blockIdx.y * 16;

    // betas tile -> LDS, zero-padded to K=352 so the K-tail reads zeros
    for (int idx = tid; idx < 16 * 352; idx += 256) {
        int m = idx / 352, l = idx - m * 352;
        float v = 0.f;
        if (l < 300)      v = sp[(b0 + m) * 300 + l];
        else if (l < LK)  v = ep[(b0 + m) * 50 + (l - 300)];
        sB[m * K1_STRIDE + l] = v;
    }
    if (tid < 32) sEl[tid >> 1][tid & 1] = eyelid[(b0 + (tid >> 1)) * 2 + (tid & 1)];
    __syncthreads();

    const int wave = tid >> 5;
    const int lane = tid & 31;
    const int m16  = lane & 15;
    const int koff = (lane >> 4) * 2;             // K-half per A/B frag layout
    const int n    = blockIdx.x * 128 + wave * 16 + m16;
    const int ncl  = (n < NV3) ? n : (NV3 - 1);   // clamp (result masked at store)
    const float* sdrow = SD + (size_t)ncl * LK;

    v8f acc = {};
    int kb = koff;
    #pragma unroll 4
    for (int kk = 0; kk < 87; ++kk) {             // 87 full K=4 steps
        v2f a  = { sB[m16 * K1_STRIDE + kb], sB[m16 * K1_STRIDE + kb + 1] };
        v2f bb = { sdrow[kb], sdrow[kb + 1] };
        acc = __builtin_amdgcn_wmma_f32_16x16x4_f32(false, a, false, bb,
                                                    (short)0, acc, false, false);
        kb += 4;
    }
    { // tail: k0=348, only k<350 valid (LDS side already zero-padded)
        v2f a = { sB[m16 * K1_STRIDE + kb], sB[m16 * K1_STRIDE + kb + 1] };
        float bv0 = (kb     < LK) ? sdrow[kb]     : 0.f;
        float bv1 = (kb + 1 < LK) ? sdrow[kb + 1] : 0.f;
        v2f bb = { bv0, bv1 };
        acc = __builtin_amdgcn_wmma_f32_16x16x4_f32(false, a, false, bb,
                                                    (short)0, acc, false, false);
    }

    const float tv = vt[ncl], lv = le[ncl], rv = re[ncl];
    const int mhi = (lane >> 4) * 8;              // C/D frag: vgpr r -> row r+mhi
    if (n < NV3) {
        #pragma unroll
        for (int r = 0; r < 8; ++r) {
            int m = r + mhi;
            vs[(size_t)(b0 + m) * NV3 + n] = acc[r] + tv + sEl[m][0] * lv + sEl[m][1] * rv;
        }
    }
}

// ===========================================================================
// Kernel 2: J[b,j,c] = sum_v Jreg[j,v] * vs[b,v,c]   (one block per batch)
// ===========================================================================
__global__ __launch_bounds__(256) void k2_jreg(
    const float* __restrict__ Jreg,   // [5,V]
    const float* __restrict__ vs,     // [B,NV3]
    float* __restrict__ Jout)         // [B,15]
{
    const int b = blockIdx.x, tid = threadIdx.x;
    float acc[15];
    #pragma unroll
    for (int i = 0; i < 15; ++i) acc[i] = 0.f;
    const float* vsb = vs + (size_t)b * NV3;
    for (int v = tid; v < V_NUM; v += 256) {
        float p0 = vsb[v * 3 + 0], p1 = vsb[v * 3 + 1], p2 = vsb[v * 3 + 2];
        #pragma unroll
        for (int j = 0; j < 5; ++j) {
            float w = Jreg[j * V_NUM + v];
            acc[j * 3 + 0] += w * p0;
            acc[j * 3 + 1] += w * p1;
            acc[j * 3 + 2] += w * p2;
        }
    }
    __shared__ float red[15][256];
    #pragma unroll
    for (int i = 0; i < 15; ++i) red[i][tid] = acc[i];
    __syncthreads();
    for (int s = 128; s > 0; s >>= 1) {
        if (tid < s) {
            #pragma unroll
            for (int i = 0; i < 15; ++i) red[i][tid] += red[i][tid + s];
        }
        __syncthreads();
    }
    if (tid < 15) Jout[b * 15 + tid] = red[tid][0];
}

// ===========================================================================
// Kernel 3: Rodrigues + kinematic chain + A matrices (one thread per batch)
// ===========================================================================
__device__ __forceinline__ void rodrigues3(float x, float y, float z, float* R) {
    float dx = x + 1e-8f, dy = y + 1e-8f, dz = z + 1e-8f;  // reference adds 1e-8 per-component
    float ang = sqrtf(dx * dx + dy * dy + dz * dz);
    float inv = 1.f / ang;
    float ax = x * inv, ay = y * inv, az = z * inv;        // axis = rvec / |rvec+eps|
    float s = sinf(ang), c = cosf(ang), t = 1.f - c;
    R[0] = 1.f - t * (ay * ay + az * az);
    R[1] = -s * az + t * ax * ay;
    R[2] =  s * ay + t * ax * az;
    R[3] =  s * az + t * ax * ay;
    R[4] = 1.f - t * (ax * ax + az * az);
    R[5] = -s * ax + t * ay * az;
    R[6] = -s * ay + t * ax * az;
    R[7] =  s * ax + t * ay * az;
    R[8] = 1.f - t * (ax * ax + ay * ay);
}

__global__ __launch_bounds__(256) void k3_chain(
    const float* __restrict__ pose, const float* __restrict__ neck,
    const float* __restrict__ jaw,  const float* __restrict__ eye,
    const float* __restrict__ Jin,  // [B,15]
    float* __restrict__ pf,         // [B,36]
    float* __restrict__ Aout)       // [B,5,12] rows 0..2 of each A
{
    int b = blockIdx.x * blockDim.x + threadIdx.x;
    if (b >= BATCH) return;
    float R[5][9];
    rodrigues3(pose[b * 3], pose[b * 3 + 1], pose[b * 3 + 2], R[0]);
    rodrigues3(neck[b * 3], neck[b * 3 + 1], neck[b * 3 + 2], R[1]);
    rodrigues3(jaw [b * 3], jaw [b * 3 + 1], jaw [b * 3 + 2], R[2]);
    rodrigues3(eye[b * 6 + 0], eye[b * 6 + 1], eye[b * 6 + 2], R[3]);
    rodrigues3(eye[b * 6 + 3], eye[b * 6 + 4], eye[b * 6 + 5], R[4]);

    #pragma unroll
    for (int j = 1; j < 5; ++j)
        #pragma unroll
        for (int i = 0; i < 9; ++i)
            pf[b * 36 + (j - 1) * 9 + i] = R[j][i] - ((i % 4 == 0) ? 1.f : 0.f);

    float J[5][3];
    #pragma unroll
    for (int j = 0; j < 5; ++j)
        #pragma unroll
        for (int c = 0; c < 3; ++c) J[j][c] = Jin[b * 15 + j * 3 + c];

    const int par[5] = {-1, 0, 1, 1, 1};
    float rel[5][3];
    #pragma unroll
    for (int c = 0; c < 3; ++c) rel[0][c] = J[0][c];
    #pragma unroll
    for (int j = 1; j < 5; ++j)
        #pragma unroll
        for (int c = 0; c < 3; ++c) rel[j][c] = J[j][c] - J[par[j]][c];

    float Gr[5][9], Gt[5][3];
    #pragma unroll
    for (int i = 0; i < 9; ++i) Gr[0][i] = R[0][i];
    #pragma unroll
    for (int c = 0; c < 3; ++c) Gt[0][c] = rel[0][c];
    #pragma unroll
    for (int j = 1; j < 5; ++j) {
        int p = par[j];
        #pragma unroll
        for (int r = 0; r < 3; ++r) {
            #pragma unroll
            for (int c = 0; c < 3; ++c)
                Gr[j][r * 3 + c] = Gr[p][r * 3 + 0] * R[j][0 + c]
                                 + Gr[p][r * 3 + 1] * R[j][3 + c]
                                 + Gr[p][r * 3 + 2] * R[j][6 + c];
            Gt[j][r] = Gr[p][r * 3 + 0] * rel[j][0] + Gr[p][r * 3 + 1] * rel[j][1]
                     + Gr[p][r * 3 + 2] * rel[j][2] + Gt[p][r];
        }
    }
    #pragma unroll
    for (int j = 0; j < 5; ++j)
        #pragma unroll
        for (int r = 0; r < 3; ++r) {
            float corr = Gr[j][r * 3 + 0] * J[j][0] + Gr[j][r * 3 + 1] * J[j][1]
                       + Gr[j][r * 3 + 2] * J[j][2];
            Aout[b * 60 + j * 12 + r * 4 + 0] = Gr[j][r * 3 + 0];
            Aout[b * 60 + j * 12 + r * 4 + 1] = Gr[j][r * 3 + 1];
            Aout[b * 60 + j * 12 + r * 4 + 2] = Gr[j][r * 3 + 2];
            Aout[b * 60 + j * 12 + r * 4 + 3] = Gt[j][r] - corr;
        }
}

// ===========================================================================
// Kernel 4: v_posed[b,n] = vs[b,n] + sum_k pf[b,k]*PD[k,n]   (K=36, WMMA)
// ===========================================================================
#define K4_STRIDE 44    // m*44 mod 64 banks -> 16 distinct, spaced by 4: conflict-free
__global__ __launch_bounds__(256) void k4_pose(
    const float* __restrict__ pf,   // [B,36]
    const float* __restrict__ PD,   // [36,NV3]
    const float* __restrict__ vs,   // [B,NV3]
    float* __restrict__ vp)         // [B,NV3]
{
    __shared__ float sP[16 * K4_STRIDE];
    const int tid = threadIdx.x;
    const int b0  = blockIdx.y * 16;
    for (int idx = tid; idx < 16 * 36; idx += 256) {
        int m = idx / 36, k = idx - m * 36;
        sP[m * K4_STRIDE + k] = pf[(b0 + m) * 36 + k];
    }
    __syncthreads();

    const int wave = tid >> 5, lane = tid & 31;
    const int m16 = lane & 15, koff = (lane >> 4) * 2;
    const int n   = blockIdx.x * 128 + wave * 16 + m16;
    const int ncl = (n < NV3) ? n : (NV3 - 1);

    v8f acc = {};
    #pragma unroll
    for (int kk = 0; kk < 9; ++kk) {     // 36 = 9 * 4, no tail
        int kb = kk * 4 + koff;
        v2f a  = { sP[m16 * K4_STRIDE + kb], sP[m16 * K4_STRIDE + kb + 1] };
        v2f bb = { PD[(size_t)kb * NV3 + ncl], PD[(size_t)(kb + 1) * NV3 + ncl] };
        acc = __builtin_amdgcn_wmma_f32_16x16x4_f32(false, a, false, bb,
                                                    (short)0, acc, false, false);
    }

    const int mhi = (lane >> 4) * 8;
    if (n < NV3) {
        #pragma unroll
        for (int r = 0; r < 8; ++r) {
            size_t off = (size_t)(b0 + r + mhi) * NV3 + n;
            vp[off] = vs[off] + acc[r];
        }
    }
}

// ===========================================================================
// Kernel 5: LBS — T = sum_j w[v,j]*A[b,j];  verts = T[:, :3] @ p + T[:,3]
// ===========================================================================
__global__ __launch_bounds__(256) void k5_lbs(
    const float* __restrict__ W,    // [V,5]
    const float* __restrict__ Aall, // [B,60]
    const float* __restrict__ vp,   // [B,NV3]
    float* __restrict__ out)        // [B,NV3]
{
    __shared__ float sA[60];
    const int b = blockIdx.y, tid = threadIdx.x;
    if (tid < 60) sA[tid] = Aall[b * 60 + tid];
    __syncthreads();
    int v = blockIdx.x * 256 + tid;
    if (v >= V_NUM) return;
    float w0 = W[v * 5 + 0], w1 = W[v * 5 + 1], w2 = W[v * 5 + 2],
          w3 = W[v * 5 + 3], w4 = W[v * 5 + 4];
    float T[12];
    #pragma unroll
    for (int i = 0; i < 12; ++i)
        T[i] = w0 * sA[i] + w1 * sA[12 + i] + w2 * sA[24 + i]
             + w3 * sA[36 + i] + w4 * sA[48 + i];
    size_t base = (size_t)b * NV3 + (size_t)v * 3;
    float px = vp[base], py = vp[base + 1], pz = vp[base + 2];
    out[base + 0] = T[0] * px + T[1] * py + T[2]  * pz + T[3];
    out[base + 1] = T[4] * px + T[5] * py + T[6]  * pz + T[7];
    out[base + 2] = T[8] * px + T[9] * py + T[10] * pz + T[11];
}

// ===========================================================================
extern "C" void kernel_launch(void* const* d_in, const int* in_sizes, int n_in,
                              void* d_out, int out_size, void* d_ws, size_t ws_size,
                              hipStream_t stream)
{
    const float* sp     = (const float*)d_in[0];   // shape_params      [B,300]
    const float* ep     = (const float*)d_in[1];   // expression_params [B,50]
    const float* pose   = (const float*)d_in[2];   // pose_params       [B,3]
    const float* jaw    = (const float*)d_in[3];   // jaw_params        [B,3]
    const float* eye    = (const float*)d_in[4];   // eye_pose_params   [B,6]
    const float* neck   = (const float*)d_in[5];   // neck_pose_params  [B,3]
    const float* eyelid = (const float*)d_in[6];   // eyelid_params     [B,2]
    const float* vt     = (const float*)d_in[7];   // v_template        [V,3]
    const float* SD     = (const float*)d_in[8];   // shapedirs         [V,3,350]
    const float* PD     = (const float*)d_in[9];   // posedirs          [36,V*3]
    const float* Jreg   = (const float*)d_in[10];  // J_regressor       [5,V]
    const float* W      = (const float*)d_in[11];  // lbs_weights       [V,5]
    const float* le     = (const float*)d_in[12];  // l_eyelid          [V,3]
    const float* re     = (const float*)d_in[13];  // r_eyelid          [V,3]
    float* verts = (float*)d_out;                  // [B,V,3]

    float* ws = (float*)d_ws;
    float* vs = ws;                                   // B*NV3
    float* vp = vs + (size_t)BATCH * NV3;             // B*NV3
    float* Jb = vp + (size_t)BATCH * NV3;             // B*15
    float* pf = Jb + (size_t)BATCH * 15;              // B*36
    float* Ab = pf + (size_t)BATCH * 36;              // B*60

    dim3 gGemm((NV3 + 127) / 128, BATCH / 16);
    k1_shape<<<gGemm, 256, 0, stream>>>(sp, ep, eyelid, vt, SD, le, re, vs);
    k2_jreg <<<BATCH, 256, 0, stream>>>(Jreg, vs, Jb);
    k3_chain<<<(BATCH + 255) / 256, 256, 0, stream>>>(pose, neck, jaw, eye, Jb, pf, Ab);
    k4_pose <<<gGemm, 256, 0, stream>>>(pf, PD, vs, vp);
    dim3 gLbs((V_NUM + 255) / 256, BATCH);
    k5_lbs  <<<gLbs, 256, 0, stream>>>(W, Ab, vp, verts);
}